// TokenProcessor_80144089743455
// MI455X (gfx1250) — compile-verified
//
#include <hip/hip_runtime.h>
#include <stdint.h>

typedef float v2f __attribute__((ext_vector_type(2)));
typedef float v8f __attribute__((ext_vector_type(8)));

#define T_STEPS 16
#define NW 8            // waves per block (wave32)
#define BLOCK (NW * 32)

__device__ __forceinline__ float fast_len(float x) {
  // v_sqrt_f32 with |src| modifier: single trans instruction, ~1ulp
  return __builtin_amdgcn_sqrtf(__builtin_fabsf(x));
}

// One block owns 16 batch rows for the entire T-step recurrence.
// dist^2 per (token, batch, corner) via V_WMMA_F32_16X16X4_F32:
//   A row (token m, corner k) = (2*fx, 2*fy, fx^2+fy^2, 1)   [precomputed in LDS]
//   B col (batch b, corner k) = (wx, wy, 1, ux^2+uy^2),  w = R^T (prev_pos - g_k)
//   D[m][b] = ||R f + u||^2
__global__ __launch_bounds__(BLOCK) void tokenize_kernel(
    const float* __restrict__ data,        // [B, 16, 3] (x, y, heading)
    const float* __restrict__ token_traj,  // [N, 4, 2]
    float* __restrict__ out,               // idx[B*T] ++ pos[B*T*2] ++ head[B*T], f32
    int B, int N)
{
  extern __shared__ char smem[];
  float2* tok  = (float2*)smem;                               // raw codebook: N*4 float2 (64KB)
  float2* frag = (float2*)(smem + (size_t)N * 4 * sizeof(float2)); // A-frags: N*8 float2 (128KB)
  char* sp = (char*)(frag + (size_t)N * 8);
  float* wxS = (float*)sp; sp += 4 * 16 * sizeof(float);
  float* wyS = (float*)sp; sp += 4 * 16 * sizeof(float);
  float* ccS = (float*)sp; sp += 4 * 16 * sizeof(float);
  float* pxS = (float*)sp; sp += 16 * sizeof(float);
  float* pyS = (float*)sp; sp += 16 * sizeof(float);
  float* phS = (float*)sp; sp += 16 * sizeof(float);
  float* redv = (float*)sp; sp += NW * 16 * sizeof(float);
  int*   redi = (int*)sp;

  const int tid  = threadIdx.x;
  const int lane = tid & 31;
  const int wave = tid >> 5;
  const int bg0  = blockIdx.x * 16;

  // ---- stage raw codebook into LDS via CDNA5 async global->LDS path ----
  {
    uint32_t ldsBase = (uint32_t)(uintptr_t)(void*)tok;
    uint64_t gbase   = (uint64_t)(uintptr_t)token_traj;
    const int iters = (N * 32) / (BLOCK * 16);               // 64KB / 4KB = 16
    for (int i = 0; i < iters; ++i) {
      uint32_t off    = (uint32_t)(i * BLOCK + tid) * 16u;
      uint32_t ldsOff = ldsBase + off;
      asm volatile("global_load_async_to_lds_b128 %0, %1, %2"
                   :: "v"(ldsOff), "v"(off), "s"(gbase) : "memory");
    }
    asm volatile("s_wait_asynccnt 0" ::: "memory");
  }
  if (tid < 16) { pxS[tid] = 0.f; pyS[tid] = 0.f; phS[tid] = 0.f; }
  __syncthreads();

  // ---- one-time: build WMMA-ready A fragments in LDS ----
  // frag[((tile*4 + k)*32 + lane)] = per-lane float2 of the 16x4 A operand
  {
    const int totalE = (N >> 4) * 128;                       // 16384
    for (int e = tid; e < totalE; e += BLOCK) {
      int lane_e = e & 31;
      int kc     = (e >> 5) & 3;
      int tile_e = e >> 7;
      float2 f = tok[(tile_e * 16 + (lane_e & 15)) * 4 + kc];
      float2 g;
      if (lane_e < 16) { g.x = 2.0f * f.x;            g.y = 2.0f * f.y; }
      else             { g.x = f.x * f.x + f.y * f.y; g.y = 1.0f; }
      frag[e] = g;
    }
  }
  __syncthreads();

  const int numTiles     = N >> 4;            // 128
  const int tilesPerWave = numTiles / NW;     // 16
  float* outIdx  = out;
  float* outPos  = out + (size_t)B * T_STEPS;
  float* outHead = out + (size_t)B * T_STEPS * 3;

  for (int t = 0; t < T_STEPS; ++t) {
    // ---- per-step setup: lane b builds B-side scalars from carry ----
    if (tid < 16) {
      int bgl = bg0 + tid;
      const float* dp = data + ((size_t)bgl * T_STEPS + t) * 3;
      float x = dp[0], y = dp[1], h = dp[2];
      float hc = 0.5f * __cosf(h), hs = 0.5f * __sinf(h);
      float lc = 4.8f * hc, ls = 4.8f * hs, wc = 2.0f * hc, ws = 2.0f * hs;
      float gx[4], gy[4];
      gx[0] = x + lc - ws; gy[0] = y + ls + wc;   // left_front
      gx[1] = x + lc + ws; gy[1] = y + ls - wc;   // right_front
      gx[2] = x - lc + ws; gy[2] = y - ls - wc;   // right_back
      gx[3] = x - lc - ws; gy[3] = y - ls + wc;   // left_back
      float ppx = pxS[tid], ppy = pyS[tid], pph = phS[tid];
      float c0 = __cosf(pph), s0 = __sinf(pph);
#pragma unroll
      for (int k = 0; k < 4; ++k) {
        float ux = ppx - gx[k], uy = ppy - gy[k];
        wxS[k * 16 + tid] = c0 * ux + s0 * uy;    // (R^T u).x
        wyS[k * 16 + tid] = -s0 * ux + c0 * uy;   // (R^T u).y
        ccS[k * 16 + tid] = ux * ux + uy * uy;
      }
    }
    __syncthreads();

    // ---- B fragments (held in VGPRs across the tile sweep) ----
    v2f Bf[4];
#pragma unroll
    for (int k = 0; k < 4; ++k) {
      float b0, b1;
      if (lane < 16) { b0 = wxS[k * 16 + lane]; b1 = wyS[k * 16 + lane]; }
      else           { b0 = 1.0f;               b1 = ccS[k * 16 + (lane - 16)]; }
      Bf[k][0] = b0; Bf[k][1] = b1;
    }

    // ---- token tile sweep: ds_load -> wmma -> sqrt/sum -> running min ----
    float minv = 3.4e38f;
    int   mini = 0;
    for (int ti = 0; ti < tilesPerWave; ++ti) {
      int tile = wave + ti * NW;
      const float2* fb = frag + (size_t)tile * 128 + lane;
      v8f D[4];
#pragma unroll
      for (int k = 0; k < 4; ++k) {
        float2 a2 = fb[k * 32];
        v2f A; A[0] = a2.x; A[1] = a2.y;
        v8f cz = {0.f, 0.f, 0.f, 0.f, 0.f, 0.f, 0.f, 0.f};
        D[k] = __builtin_amdgcn_wmma_f32_16x16x4_f32(
            false, A, false, Bf[k], (short)0, cz, false, false);
      }
      int mbase = tile * 16 + ((lane >> 4) << 3);
#pragma unroll
      for (int r = 0; r < 8; ++r) {
        float dsum = (fast_len(D[0][r]) + fast_len(D[1][r])) +
                     (fast_len(D[2][r]) + fast_len(D[3][r]));
        // indices strictly increase within a lane's scan -> strict < keeps first min
        if (dsum < minv) { minv = dsum; mini = mbase + r; }
      }
    }

    // ---- argmin reduction: hi half-lanes -> lo, then across waves via LDS ----
    {
      float v2 = __shfl_down(minv, 16);
      int   i2 = __shfl_down(mini, 16);
      if (lane < 16) {
        if (v2 < minv || (v2 == minv && i2 < mini)) { minv = v2; mini = i2; }
        redv[wave * 16 + lane] = minv;
        redi[wave * 16 + lane] = mini;
      }
    }
    __syncthreads();

    // ---- finalize: selected contour -> new pose, write outputs, update carry ----
    if (tid < 16) {
      float bv = redv[lane]; int bi = redi[lane];
#pragma unroll
      for (int w = 1; w < NW; ++w) {
        float v = redv[w * 16 + lane]; int i = redi[w * 16 + lane];
        if (v < bv || (v == bv && i < bi)) { bv = v; bi = i; }
      }
      float ppx = pxS[lane], ppy = pyS[lane], pph = phS[lane];
      float c0 = __cosf(pph), s0 = __sinf(pph);
      float sx = 0.f, sy = 0.f, x0 = 0.f, y0 = 0.f, x3 = 0.f, y3 = 0.f;
#pragma unroll
      for (int k = 0; k < 4; ++k) {
        float2 f = tok[bi * 4 + k];
        float tx = c0 * f.x - s0 * f.y + ppx;
        float ty = s0 * f.x + c0 * f.y + ppy;
        sx += tx; sy += ty;
        if (k == 0) { x0 = tx; y0 = ty; }
        if (k == 3) { x3 = tx; y3 = ty; }
      }
      float nx = sx * 0.25f, ny = sy * 0.25f;
      float nh = atan2f(y0 - y3, x0 - x3);
      int bgl = bg0 + lane;
      size_t bt = (size_t)bgl * T_STEPS + t;
      outIdx[bt]          = (float)bi;
      outPos[bt * 2]      = nx;
      outPos[bt * 2 + 1]  = ny;
      outHead[bt]         = nh;
      pxS[lane] = nx; pyS[lane] = ny; phS[lane] = nh;
    }
    __syncthreads();
  }
}

extern "C" void kernel_launch(void* const* d_in, const int* in_sizes, int n_in,
                              void* d_out, int out_size, void* d_ws, size_t ws_size,
                              hipStream_t stream) {
  (void)n_in; (void)out_size; (void)d_ws; (void)ws_size;
  const float* data = (const float*)d_in[0];      // [B,16,3]
  const float* tokg = (const float*)d_in[1];      // [N,4,2]
  float* out = (float*)d_out;
  int B = in_sizes[0] / (T_STEPS * 3);            // 4096
  int N = in_sizes[1] / 8;                        // 2048
  size_t shmem = (size_t)N * 4 * sizeof(float2)   // raw codebook 64KB
               + (size_t)N * 8 * sizeof(float2)   // A fragments 128KB
               + (4 * 16 * 3 + 16 * 3) * sizeof(float)
               + (size_t)NW * 16 * (sizeof(float) + sizeof(int));
  dim3 grid(B / 16), block(BLOCK);
  tokenize_kernel<<<grid, block, shmem, stream>>>(data, tokg, out, B, N);
}